// ActiveBoundaryLoss_54924041781920
// MI455X (gfx1250) — compile-verified
//
#include <hip/hip_runtime.h>
#include <hip/hip_bf16.h>
#include <math.h>
#include <stdint.h>

// Problem shape (from setup_inputs): inp [1,2,48,160,160] f32, target [1,48,160,160] i32
#define ABL_D 48
#define ABL_H 160
#define ABL_W 160
#define ABL_HW (ABL_H * ABL_W)
#define ABL_N (ABL_D * ABL_HW)
#define ABL_THETA 20.0f
#define ABL_K4_BLOCKS 1200

#define ABL_AS1 __attribute__((address_space(1)))
#define ABL_AS3 __attribute__((address_space(3)))

typedef float v2f __attribute__((ext_vector_type(2)));
typedef float v8f __attribute__((ext_vector_type(8)));

// ---------------- CDNA5 async global->LDS path (guarded) ----------------
#if __has_builtin(__builtin_amdgcn_global_load_async_to_lds_b32)
#define ABL_ASYNC_BUILTIN 1
#else
#define ABL_ASYNC_BUILTIN 0
#endif

__device__ __forceinline__ void abl_async_copy_b32(const float* g, float* l) {
#if ABL_ASYNC_BUILTIN
  __builtin_amdgcn_global_load_async_to_lds_b32(
      (ABL_AS1 int*)g, (ABL_AS3 int*)l, 0, 0);
#else
  // ISA 15.18.3 opcode 96: GLOBAL_LOAD_ASYNC_TO_LDS_B32 vdst(lds addr), vaddr(64b), off
  asm volatile("global_load_async_to_lds_b32 %0, %1, off"
               :: "v"((unsigned)(uintptr_t)l),
                  "v"((unsigned long long)(uintptr_t)g)
               : "memory");
#endif
}

__device__ __forceinline__ void abl_wait_async() {
#if __has_builtin(__builtin_amdgcn_s_wait_asynccnt)
  __builtin_amdgcn_s_wait_asynccnt(0);
#else
  asm volatile("s_wait_asynccnt 0" ::: "memory");
#endif
}

// ---------------- helpers ----------------
__device__ __forceinline__ unsigned abl_f2key(float f) {
  unsigned u = __float_as_uint(f);
  return (u & 0x80000000u) ? ~u : (u | 0x80000000u);
}
__device__ __forceinline__ float abl_key2f(unsigned k) {
  unsigned u = (k & 0x80000000u) ? (k & 0x7fffffffu) : ~k;
  return __uint_as_float(u);
}
// mean over C=2 of KLDivLoss(inp=p(center), tgt=t(neighbor)) = xlogy(t,t) - t*p
__device__ __forceinline__ float abl_kldm(float p0, float p1, float t0, float t1) {
  float a0 = (t0 > 0.f) ? t0 * logf(t0) : 0.f;
  float a1 = (t1 > 0.f) ? t1 * logf(t1) : 0.f;
  return 0.5f * ((a0 - t0 * p0) + (a1 - t1 * p1));
}

// ---------------- K1: kl_vals + gdb seed ----------------
__global__ void k_prepare(const float* __restrict__ inp, const int* __restrict__ tgt,
                          float* __restrict__ klvals, float* __restrict__ fA) {
  for (int idx = blockIdx.x * blockDim.x + threadIdx.x; idx < ABL_N;
       idx += gridDim.x * blockDim.x) {
    __builtin_prefetch(inp + idx + 4096, 0, 1);          // global_prefetch_b8
    __builtin_prefetch(inp + ABL_N + idx + 4096, 0, 1);
    int w = idx % ABL_W;
    int h = (idx / ABL_W) % ABL_H;
    int d = idx / ABL_HW;
    float p0 = inp[idx], p1 = inp[ABL_N + idx];
    float kh, kv, kd;
    if (w == ABL_W - 1) kh = -1.f;
    else kh = abl_kldm(p0, p1, inp[idx + 1], inp[ABL_N + idx + 1]);
    if (h == ABL_H - 1) kv = -1.f;
    else kv = abl_kldm(p0, p1, inp[idx + ABL_W], inp[ABL_N + idx + ABL_W]);
    if (d == ABL_D - 1) kd = -1.f;
    else kd = abl_kldm(p0, p1, inp[idx + ABL_HW], inp[ABL_N + idx + ABL_HW]);
    klvals[idx] = fmaxf(fmaxf(kh, kv), kd);

    int t  = tgt[idx];
    int th = (w < ABL_W - 1) ? tgt[idx + 1] : 0;
    int tv = (h < ABL_H - 1) ? tgt[idx + ABL_W] : 0;
    int td = (d < ABL_D - 1) ? tgt[idx + ABL_HW] : 0;
    bool gdb = (3 * t) != (th + tv + td);
    fA[idx] = gdb ? 0.f : 1e10f;
  }
}

// ---------------- radix select (exact quantile) ----------------
__global__ void k_sel_init(unsigned* __restrict__ hist, unsigned* __restrict__ prefix,
                           unsigned* __restrict__ rank) {
  int t = blockIdx.x * blockDim.x + threadIdx.x;
  if (t < 512) hist[t] = 0;
  if (t == 0) {
    prefix[0] = 0; prefix[1] = 0;
    // pos = 0.99*(N-1) = 1216511.01 -> ranks 1216511 and 1216512 (0-indexed asc)
    rank[0] = 1216511u; rank[1] = 1216512u;
  }
}

__global__ void k_sel_hist(const float* __restrict__ klvals, unsigned* __restrict__ hist,
                           const unsigned* __restrict__ prefix, int pass) {
  __shared__ unsigned lh[512];
  for (int i = threadIdx.x; i < 512; i += blockDim.x) lh[i] = 0;
  __syncthreads();
  unsigned shift = (unsigned)pass * 8u;
  unsigned maskhi = (pass == 3) ? 0u : (0xFFFFFFFFu << (shift + 8u));
  unsigned p0 = prefix[0] & maskhi, p1 = prefix[1] & maskhi;
  for (int idx = blockIdx.x * blockDim.x + threadIdx.x; idx < ABL_N;
       idx += gridDim.x * blockDim.x) {
    unsigned k = abl_f2key(klvals[idx]);
    unsigned khi = k & maskhi;
    unsigned b = (k >> shift) & 0xFFu;
    if (khi == p0) atomicAdd(&lh[b], 1u);
    if (khi == p1) atomicAdd(&lh[256 + b], 1u);
  }
  __syncthreads();
  for (int i = threadIdx.x; i < 512; i += blockDim.x)
    if (lh[i]) atomicAdd(&hist[i], lh[i]);
}

__global__ void k_sel_scan(unsigned* __restrict__ hist, unsigned* __restrict__ prefix,
                           unsigned* __restrict__ rank, float* __restrict__ thr, int pass) {
  if (threadIdx.x == 0) {
    unsigned shift = (unsigned)pass * 8u;
    for (int s = 0; s < 2; ++s) {
      unsigned r = rank[s], c = 0, chosen = 255u;
      for (int b = 0; b < 256; ++b) {
        unsigned hh = hist[s * 256 + b];
        if (c + hh > r) { chosen = (unsigned)b; r -= c; break; }
        c += hh;
      }
      rank[s] = r;
      prefix[s] |= chosen << shift;
    }
    if (pass == 0) {
      double v0 = (double)abl_key2f(prefix[0]);
      double v1 = (double)abl_key2f(prefix[1]);
      double pos = 0.99 * (double)(ABL_N - 1);
      double frac = pos - floor(pos);
      *thr = (float)(v0 + frac * (v1 - v0));
    }
  }
  __syncthreads();
  for (int i = threadIdx.x; i < 512; i += blockDim.x) hist[i] = 0;  // for next pass
}

// ---------------- EDT: 1D min-plus DT per line, LDS-staged via async copy ----------------
__global__ void k_edt(const float* __restrict__ in, float* __restrict__ out,
                      int axis, int doSqrt) {
  __shared__ float line[ABL_H];  // max line length 160
  int L, S;
  long long base;
  int li = blockIdx.x;
  if (axis == 0)      { L = ABL_D; S = ABL_HW; base = li; }                 // lines along D
  else if (axis == 1) { L = ABL_H; S = ABL_W;                               // lines along H
                        int d = li / ABL_W, w = li % ABL_W;
                        base = (long long)d * ABL_HW + w; }
  else                { L = ABL_W; S = 1; base = (long long)li * ABL_W; }   // lines along W
  int t = threadIdx.x;
  if (t < L) abl_async_copy_b32(in + base + (long long)t * S, &line[t]);
  abl_wait_async();
  __syncthreads();
  if (t < L) {
    float ti = (float)t;
    float m = 3.0e38f;
    for (int j = 0; j < L; ++j) {
      float dj = ti - (float)j;
      m = fminf(m, line[j] + dj * dj);
    }
    out[base + (long long)t * S] = doSqrt ? sqrtf(m) : m;
  }
}

// ---------------- K4: 26-direction loss, pdb-masked, per-block partials ----------------
__global__ void k_loss(const float* __restrict__ inp, const float* __restrict__ klvals,
                       const float* __restrict__ dist, const float* __restrict__ thrp,
                       float* __restrict__ pLoss, unsigned* __restrict__ pCnt) {
  const float thr = *thrp;
  float accL = 0.f;
  unsigned accC = 0u;
  for (int idx = blockIdx.x * blockDim.x + threadIdx.x; idx < ABL_N;
       idx += gridDim.x * blockDim.x) {
    __builtin_prefetch(klvals + idx + 4096, 0, 1);
    bool pdb = klvals[idx] >= thr;
    if (!pdb) continue;
    ++accC;
    float gd = dist[idx];
    if (gd == 0.f) continue;

    int w = idx % ABL_W;
    int h = (idx / ABL_W) % ABL_H;
    int d = idx / ABL_HW;
    float p0 = inp[idx], p1 = inp[ABL_N + idx];

    float dpi[26], ds26[26];
    int dir = 0;
    float s = 0.f;
    for (int i = -1; i <= 1; ++i)
      for (int j = -1; j <= 1; ++j)
        for (int k = -1; k <= 1; ++k) {
          if (i == 0 && j == 0 && k == 0) continue;
          int nd = d + i, nh = h + j, nw = w + k;
          bool inb = (nd >= 0 && nd < ABL_D && nh >= 0 && nh < ABL_H &&
                      nw >= 0 && nw < ABL_W);
          long long nidx = (long long)nd * ABL_HW + (long long)nh * ABL_W + nw;
          ds26[dir] = inb ? dist[nidx] : 0.f;  // zero-padded shift of gdb_dist
          // faithful reproduction of the reference's (batch,d,h) coordinate masking:
          // i==+1 -> require batch==D-1 (never, B==1); j==+1 -> require d==H-1 (never, D<H)
          // j==-1 -> require d==0; k==+1 -> require h==W-1; k==-1 -> require h==0
          bool zeroed = (i == 1) || (j == 1) ||
                        (j == -1 && d != 0) ||
                        (k == 1 && h != ABL_W - 1) ||
                        (k == -1 && h != 0);
          float v = 0.f;
          if (!zeroed) {
            float km = 0.f;  // zero-padded neighbor -> kld==0 -> exp==1
            if (inb) km = abl_kldm(p0, p1, inp[nidx], inp[ABL_N + nidx]);
            v = expf(km);
          }
          dpi[dir] = v;
          s += v;
          ++dir;
        }
    float dn = (s == 0.f) ? 1.f : s;
    int mi = 0;
    float mv = ds26[0];
    for (int q = 1; q < 26; ++q)
      if (ds26[q] < mv) { mv = ds26[q]; mi = q; }  // first min, like torch.argmin
    float bs = 0.f;
    for (int q = 0; q < 26; ++q) {
      float x = dpi[q] / dn;
      float y = (q == mi) ? 0.8f : (0.2f / 26.f);
      bs += fmaxf(x, 0.f) - x * y + log1pf(expf(-fabsf(x)));
    }
    float weight = fminf(gd, ABL_THETA) / ABL_THETA;
    accL += weight * (bs / 26.f);
  }
  __shared__ float sl[256];
  __shared__ unsigned sc[256];
  sl[threadIdx.x] = accL;
  sc[threadIdx.x] = accC;
  __syncthreads();
  for (int off = 128; off > 0; off >>= 1) {
    if ((int)threadIdx.x < off) {
      sl[threadIdx.x] += sl[threadIdx.x + off];
      sc[threadIdx.x] += sc[threadIdx.x + off];
    }
    __syncthreads();
  }
  if (threadIdx.x == 0) { pLoss[blockIdx.x] = sl[0]; pCnt[blockIdx.x] = sc[0]; }
}

// ---------------- K5: final reduction; loss sum via V_WMMA_F32_16X16X4_F32 ----------------
// Sum of K partials as a matrix product: A (16x4) holds the 32 per-lane partials
// (a = {partial, 0} per lane fills every A slot exactly once under any layout
// bijection), B = all-ones (layout invariant) => D[m,n] = sum_k A[m,k].
// With the documented C/D layout (lane 0: N=0, VGPR r = M=r; lane 16: M=8+r),
// total = sum(c[0..7] @ lane0) + sum(c[0..7] @ lane16).
__global__ void k_final(const float* __restrict__ pLoss, const unsigned* __restrict__ pCnt,
                        float* __restrict__ out) {
  int l = threadIdx.x;  // single wave of 32
  float a = 0.f;
  unsigned c = 0u;
  for (int i = l; i < ABL_K4_BLOCKS; i += 32) { a += pLoss[i]; c += pCnt[i]; }
  // deterministic integer wave reduction for n_pdb
  for (int off = 16; off > 0; off >>= 1) c += __shfl_xor(c, off, 32);
  // f32 WMMA reduction for the loss sum (EXEC all-ones: full wave, no divergence)
  v2f av; av.x = a;   av.y = 0.f;
  v2f bv; bv.x = 1.f; bv.y = 1.f;
  v8f cv = {};
  cv = __builtin_amdgcn_wmma_f32_16x16x4_f32(
      /*neg_a=*/false, av, /*neg_b=*/false, bv,
      /*c_mod=*/(short)0, cv, /*reuse_a=*/false, /*reuse_b=*/false);
  float part = cv[0] + cv[1] + cv[2] + cv[3] + cv[4] + cv[5] + cv[6] + cv[7];
  float tot = part + __shfl(part, 16, 32);  // lane0: M=0..7, lane16: M=8..15 (col 0)
  if (l == 0) out[0] = tot / (float)c;
}

// ---------------- launch ----------------
extern "C" void kernel_launch(void* const* d_in, const int* in_sizes, int n_in,
                              void* d_out, int out_size, void* d_ws, size_t ws_size,
                              hipStream_t stream) {
  const float* inp = (const float*)d_in[0];   // [2,48,160,160]
  const int* tgt   = (const int*)d_in[1];     // [48,160,160]
  float* out = (float*)d_out;

  char* ws = (char*)d_ws;
  float* klvals   = (float*)(ws);
  float* fA       = (float*)(ws + 4ull * ABL_N);
  float* fB       = (float*)(ws + 8ull * ABL_N);
  unsigned* hist  = (unsigned*)(ws + 12ull * ABL_N);
  unsigned* prefix = hist + 512;
  unsigned* rank   = prefix + 2;
  float* thr       = (float*)(rank + 2);
  float* pLoss     = (float*)(ws + 12ull * ABL_N + 4096);
  unsigned* pCnt   = (unsigned*)(pLoss + ABL_K4_BLOCKS);

  k_prepare<<<ABL_N / 256, 256, 0, stream>>>(inp, tgt, klvals, fA);

  k_sel_init<<<2, 256, 0, stream>>>(hist, prefix, rank);
  for (int pass = 3; pass >= 0; --pass) {
    k_sel_hist<<<512, 256, 0, stream>>>(klvals, hist, prefix, pass);
    k_sel_scan<<<1, 256, 0, stream>>>(hist, prefix, rank, thr, pass);
  }

  // EDT passes in reference order D,H,W; sqrt fused into last pass
  k_edt<<<ABL_HW, 64, 0, stream>>>(fA, fB, 0, 0);
  k_edt<<<ABL_D * ABL_W, 160, 0, stream>>>(fB, fA, 1, 0);
  k_edt<<<ABL_D * ABL_H, 160, 0, stream>>>(fA, fB, 2, 1);

  k_loss<<<ABL_K4_BLOCKS, 256, 0, stream>>>(inp, klvals, fB, thr, pLoss, pCnt);
  k_final<<<1, 32, 0, stream>>>(pLoss, pCnt, out);
}